// RasterMamba_27943057228196
// MI455X (gfx1250) — compile-verified
//
#include <hip/hip_runtime.h>
#include <hip/hip_bf16.h>
#include <math.h>

// ---------------------------------------------------------------------------
// RasterMamba on MI455X (gfx1250, wave32).
// B=2, D_MODEL=D_INNER=256, L=4096, D_STATE=16, DT_RANK=16, D_CONV=4.
// GEMMs: native fp32 WMMA V_WMMA_F32_16X16X4_F32, register-blocked 4 N-tiles
// per wave (A fragment reused 4x per K-step). Scan: 1 lane per (b,d,s),
// shfl_xor reduction over 16 states, gating fused into the scan's store.
// ---------------------------------------------------------------------------

typedef __attribute__((ext_vector_type(2))) float v2f;
typedef __attribute__((ext_vector_type(8))) float v8f;

#define BB     2
#define LL     4096
#define DD     256      // D_MODEL == D_INNER
#define NIN    512      // 2*D_INNER
#define NG     48       // dt_rank + 2*d_state
#define NS     16       // d_state
#define NR     16       // dt_rank

__device__ __forceinline__ v8f wmma_f32_k4(v2f a, v2f b, v8f c) {
    // 8 args: (neg_a, A, neg_b, B, c_mod, C, reuse_a, reuse_b)
    return __builtin_amdgcn_wmma_f32_16x16x4_f32(false, a, false, b,
                                                 (short)0, c, false, false);
}

__device__ __forceinline__ float sigmoidf_(float v) {
    return 1.0f / (1.0f + expf(-v));
}

// ---------------------------------------------------------------------------
// K1: P[b,l,0:512] = seq[b,l,:] @ W_in ; seq[l,c] = x[b,c,l] (A is strided).
// grid (L/16, 1, B), block 256: 8 waves x 4 N-tiles = all 512 columns.
// ---------------------------------------------------------------------------
__global__ void k_inproj(const float* __restrict__ x,
                         const float* __restrict__ Win,
                         float* __restrict__ P) {
    const int lane = threadIdx.x & 31;
    const int wave = threadIdx.x >> 5;   // 0..7
    const int hp   = lane >> 4;
    const int m    = lane & 15;
    const int l0   = blockIdx.x * 16;
    const int nb   = wave * 64;          // 4 tiles of 16
    const int b    = blockIdx.z;

    const float* xb = x + (size_t)b * DD * LL;
    v8f acc[4] = {};
    for (int k0 = 0; k0 < DD; k0 += 4) {
        const int ka = k0 + 2 * hp;
        v2f a;
        a.x = xb[(size_t)(ka + 0) * LL + l0 + m];
        a.y = xb[(size_t)(ka + 1) * LL + l0 + m];
        #pragma unroll
        for (int t = 0; t < 4; ++t) {
            v2f w;
            w.x = Win[(size_t)(ka + 0) * NIN + nb + 16 * t + m];
            w.y = Win[(size_t)(ka + 1) * NIN + nb + 16 * t + m];
            acc[t] = wmma_f32_k4(a, w, acc[t]);
        }
    }
    float* Pb = P + ((size_t)b * LL + l0) * NIN;
    #pragma unroll
    for (int t = 0; t < 4; ++t)
        #pragma unroll
        for (int r = 0; r < 8; ++r)
            Pb[(size_t)(r + 8 * hp) * NIN + nb + 16 * t + m] = acc[t][r];
}

// ---------------------------------------------------------------------------
// K2: causal depthwise conv (k=4) over flattened L + SiLU -> xc[b,l,d].
// ---------------------------------------------------------------------------
__global__ void k_conv_silu(const float* __restrict__ P,
                            const float* __restrict__ cw,
                            const float* __restrict__ cb,
                            float* __restrict__ xc) {
    const int idx = blockIdx.x * blockDim.x + threadIdx.x;  // B*L*D threads
    const int d = idx & (DD - 1);
    const int r = idx >> 8;              // b*L + l
    const int l = r & (LL - 1);
    const int b = r >> 12;
    float acc = cb[d];
    #pragma unroll
    for (int k = 0; k < 4; ++k) {
        const int ll = l + k - 3;
        if (ll >= 0)
            acc += P[((size_t)(b * LL + ll)) * NIN + d] * cw[d * 4 + k];
    }
    xc[idx] = acc * sigmoidf_(acc);      // SiLU
}

// ---------------------------------------------------------------------------
// K3: dbc[b,l,0:48] = xc[b,l,:] @ W_x.
// grid (L/16, 1, B), block 32: one wave, 3 N-tiles (48 cols), A reused 3x.
// ---------------------------------------------------------------------------
__global__ void k_wx(const float* __restrict__ xc,
                     const float* __restrict__ Wx,
                     float* __restrict__ dbc) {
    const int lane = threadIdx.x & 31;
    const int hp   = lane >> 4;
    const int m    = lane & 15;
    const int l0   = blockIdx.x * 16;
    const int b    = blockIdx.z;

    const float* A = xc + ((size_t)b * LL + l0) * DD;  // row-major [16][256]
    v8f acc[3] = {};
    for (int k0 = 0; k0 < DD; k0 += 4) {
        const int ka = k0 + 2 * hp;
        const v2f a = *(const v2f*)(A + (size_t)m * DD + ka);  // 8B aligned
        #pragma unroll
        for (int t = 0; t < 3; ++t) {
            v2f w;
            w.x = Wx[(size_t)(ka + 0) * NG + 16 * t + m];
            w.y = Wx[(size_t)(ka + 1) * NG + 16 * t + m];
            acc[t] = wmma_f32_k4(a, w, acc[t]);
        }
    }
    float* O = dbc + ((size_t)b * LL + l0) * NG;
    #pragma unroll
    for (int t = 0; t < 3; ++t)
        #pragma unroll
        for (int r = 0; r < 8; ++r)
            O[(size_t)(r + 8 * hp) * NG + 16 * t + m] = acc[t][r];
}

// ---------------------------------------------------------------------------
// K4: dt[b,l,d] = softplus(dbc[b,l,0:16] @ W_dt + b_dt).
// grid (L/16, 1, B), block 128: 4 waves x 4 N-tiles = all 256 columns.
// ---------------------------------------------------------------------------
__global__ void k_dt(const float* __restrict__ dbc,
                     const float* __restrict__ Wdt,
                     const float* __restrict__ bdt,
                     float* __restrict__ dt) {
    const int lane = threadIdx.x & 31;
    const int wave = threadIdx.x >> 5;   // 0..3
    const int hp   = lane >> 4;
    const int m    = lane & 15;
    const int l0   = blockIdx.x * 16;
    const int nb   = wave * 64;
    const int b    = blockIdx.z;

    const float* A = dbc + ((size_t)b * LL + l0) * NG;
    v8f acc[4] = {};
    #pragma unroll
    for (int k0 = 0; k0 < NR; k0 += 4) {
        const int ka = k0 + 2 * hp;
        const v2f a = *(const v2f*)(A + (size_t)m * NG + ka);  // 8B aligned
        #pragma unroll
        for (int t = 0; t < 4; ++t) {
            v2f w;
            w.x = Wdt[(size_t)(ka + 0) * DD + nb + 16 * t + m];
            w.y = Wdt[(size_t)(ka + 1) * DD + nb + 16 * t + m];
            acc[t] = wmma_f32_k4(a, w, acc[t]);
        }
    }
    float* O = dt + ((size_t)b * LL + l0) * DD;
    #pragma unroll
    for (int t = 0; t < 4; ++t) {
        const float bias = bdt[nb + 16 * t + m];
        #pragma unroll
        for (int r = 0; r < 8; ++r) {
            float v = acc[t][r] + bias;
            v = (v > 20.0f) ? v : log1pf(expf(v));   // softplus
            O[(size_t)(r + 8 * hp) * DD + nb + 16 * t + m] = v;
        }
    }
}

// ---------------------------------------------------------------------------
// K5: selective scan + fused gate. One lane per (b,d,s); 2 channels/wave.
// grid (B*D/2) blocks of 32.  g[b,l,d] = (<h,C> + xc*Dvec) * silu(z).
// ---------------------------------------------------------------------------
__global__ void k_scan_gate(const float* __restrict__ dt,
                            const float* __restrict__ dbc,
                            const float* __restrict__ xc,
                            const float* __restrict__ A_log,
                            const float* __restrict__ Dvec,
                            const float* __restrict__ P,   // z = P[:,:,256:512]
                            float* __restrict__ g) {
    const int lane = threadIdx.x;        // 0..31
    const int hp   = lane >> 4;
    const int s    = lane & 15;
    const int ch   = blockIdx.x * 2 + hp;   // 0..511
    const int b    = ch >> 8;
    const int d    = ch & (DD - 1);

    const float Ads = -expf(A_log[d * NS + s]);
    const float Dd  = Dvec[d];
    const size_t baseX = (size_t)b * LL * DD + d;
    const size_t baseG = (size_t)b * LL * NG;
    const size_t baseZ = (size_t)b * LL * NIN + DD + d;

    float h = 0.0f;
    for (int l = 0; l < LL; ++l) {
        const float dtv = dt [baseX + (size_t)l * DD];
        const float xv  = xc [baseX + (size_t)l * DD];
        const float Bv  = dbc[baseG + (size_t)l * NG + NR + s];
        const float Cv  = dbc[baseG + (size_t)l * NG + NR + NS + s];
        h = h * __expf(dtv * Ads) + (dtv * xv) * Bv;
        float p = h * Cv;
        p += __shfl_xor(p, 1, 32);       // reduce over 16 states
        p += __shfl_xor(p, 2, 32);
        p += __shfl_xor(p, 4, 32);
        p += __shfl_xor(p, 8, 32);
        if (s == 0) {
            const float z = P[baseZ + (size_t)l * NIN];
            g[baseX + (size_t)l * DD] = (p + xv * Dd) * (z * sigmoidf_(z));
        }
    }
}

// ---------------------------------------------------------------------------
// K6: out[b,c,l] = (g[b,l,:] @ W_out)[c]  (store transposed to (B,C,H,W)).
// grid (L/16, 1, B), block 128: 4 waves x 4 N-tiles = all 256 columns.
// ---------------------------------------------------------------------------
__global__ void k_outproj(const float* __restrict__ g,
                          const float* __restrict__ Wout,
                          float* __restrict__ out) {
    const int lane = threadIdx.x & 31;
    const int wave = threadIdx.x >> 5;   // 0..3
    const int hp   = lane >> 4;
    const int m    = lane & 15;
    const int l0   = blockIdx.x * 16;
    const int nb   = wave * 64;
    const int b    = blockIdx.z;

    const float* A = g + ((size_t)b * LL + l0) * DD;
    v8f acc[4] = {};
    for (int k0 = 0; k0 < DD; k0 += 4) {
        const int ka = k0 + 2 * hp;
        const v2f a = *(const v2f*)(A + (size_t)m * DD + ka);  // 8B aligned
        #pragma unroll
        for (int t = 0; t < 4; ++t) {
            v2f w;
            w.x = Wout[(size_t)(ka + 0) * DD + nb + 16 * t + m];
            w.y = Wout[(size_t)(ka + 1) * DD + nb + 16 * t + m];
            acc[t] = wmma_f32_k4(a, w, acc[t]);
        }
    }
    // out[(b*256 + n)*L + l0 + M], M = r + 8*hp: 8 consecutive floats per lane
    #pragma unroll
    for (int t = 0; t < 4; ++t) {
        float* O = out + ((size_t)b * DD + nb + 16 * t + m) * LL + l0 + 8 * hp;
        #pragma unroll
        for (int r = 0; r < 8; ++r)
            O[r] = acc[t][r];
    }
}

// ---------------------------------------------------------------------------
extern "C" void kernel_launch(void* const* d_in, const int* in_sizes, int n_in,
                              void* d_out, int out_size, void* d_ws, size_t ws_size,
                              hipStream_t stream) {
    const float* x      = (const float*)d_in[0];   // (2,256,64,64)
    const float* W_in   = (const float*)d_in[1];   // (256,512)
    const float* conv_w = (const float*)d_in[2];   // (256,4)
    const float* conv_b = (const float*)d_in[3];   // (256,)
    const float* W_x    = (const float*)d_in[4];   // (256,48)
    const float* W_dt   = (const float*)d_in[5];   // (16,256)
    const float* b_dt   = (const float*)d_in[6];   // (256,)
    const float* A_log  = (const float*)d_in[7];   // (256,16)
    const float* Dvec   = (const float*)d_in[8];   // (256,)
    const float* W_out  = (const float*)d_in[9];   // (256,256)
    float* out = (float*)d_out;                    // (2,256,64,64)

    float* ws  = (float*)d_ws;
    float* P   = ws;                               // B*L*512
    float* xc  = P   + (size_t)BB * LL * NIN;      // B*L*256
    float* dbc = xc  + (size_t)BB * LL * DD;       // B*L*48
    float* dt  = dbc + (size_t)BB * LL * NG;       // B*L*256
    float* g   = dt  + (size_t)BB * LL * DD;       // B*L*256

    // 1) in-projection (WMMA f32, 4 N-tiles/wave)
    k_inproj<<<dim3(LL / 16, 1, BB), 256, 0, stream>>>(x, W_in, P);
    // 2) causal conv + SiLU
    k_conv_silu<<<dim3((BB * LL * DD) / 256), 256, 0, stream>>>(P, conv_w, conv_b, xc);
    // 3) x-projection -> (dt_raw | B | C)  (WMMA f32, 3 N-tiles/wave)
    k_wx<<<dim3(LL / 16, 1, BB), 32, 0, stream>>>(xc, W_x, dbc);
    // 4) dt = softplus(low-rank proj)  (WMMA f32, 4 N-tiles/wave)
    k_dt<<<dim3(LL / 16, 1, BB), 128, 0, stream>>>(dbc, W_dt, b_dt, dt);
    // 5) selective scan (serial in L, parallel over b,d,s) + fused gating
    k_scan_gate<<<dim3((BB * DD) / 2), 32, 0, stream>>>(dt, dbc, xc, A_log,
                                                        Dvec, P, g);
    // 6) out-projection + transpose store (WMMA f32, 4 N-tiles/wave)
    k_outproj<<<dim3(LL / 16, 1, BB), 128, 0, stream>>>(g, W_out, out);
}